// Net_LSV_6708738916358
// MI455X (gfx1250) — compile-verified
//
#include <hip/hip_runtime.h>
#include <hip/hip_bf16.h>

typedef float v2f __attribute__((ext_vector_type(2)));
typedef float v8f __attribute__((ext_vector_type(8)));

#define MC 16384
#define NT 96
#define NSTK 21
#define NMAT 4
#define PATHS_PER_BLK 64
#define THREADS 256
#define NBLK (MC / PATHS_PER_BLK) /* 256 */

// ---- LDS layout (float offsets) ----
#define OFF_W2 0       /* 100 x 112 padded = 11200 */
#define OFF_H1 11200   /* 64 x 104 = 6656 (reused as price buffer at maturity) */
#define OFF_H2 17856   /* 64 x 112 = 7168 */
#define OFF_SW1 25024  /* 3 x 100 */
#define OFF_SB1 25324  /* 100 */
#define OFF_SW3 25424  /* 100 */
#define OFF_SB2 25524  /* 112 padded */
#define OFF_TG 25636   /* 97 */
#define OFF_STR 25733  /* 21 */
#define OFF_VDW1 25754 /* 20 */
#define OFF_VDB1 25774
#define OFF_VDW2 25794
#define OFF_VVW1 25814
#define OFF_VVB1 25834
#define OFF_VVW2 25854
#define OFF_VHW1 25874 /* 4x2x20 = 160 */
#define OFF_VHB1 26034 /* 80 */
#define OFF_VHW2 26114 /* 4x20x21 = 1680 */
#define OFF_VHB2 27794 /* 84 */
#define SMEM_FLOATS 27880

struct P {
  const float *S0, *rate, *z, *zz, *tg, *strikes, *v0, *rho;
  const float *svW1, *svb1, *svW2, *svb2, *svW3, *svb3;
  const float *vhW1, *vhb1, *vhW2, *vhb2;
  const float *vdW1, *vdb1, *vdW2, *vdb2;
  const float *vvW1, *vvb1, *vvW2, *vvb2;
  const int* mats;
  float *psum, *psq; /* each [NMAT][NBLK][NSTK] */
};

__device__ __forceinline__ float softplusf(float x) {
  return fmaxf(x, 0.f) + log1pf(expf(-fabsf(x)));
}

__global__ __launch_bounds__(THREADS) void lsv_sim(P p) {
  extern __shared__ float sm[];
  const int tid = threadIdx.x;
  const int bid = blockIdx.x;

  // ---- one-time preload of weights into LDS ----
  for (int i = tid; i < 100 * 112; i += THREADS) {
    int k = i / 112, n = i % 112;
    sm[OFF_W2 + i] = (n < 100) ? p.svW2[k * 100 + n] : 0.f;
  }
  for (int i = tid; i < 300; i += THREADS) sm[OFF_SW1 + i] = p.svW1[i];
  for (int i = tid; i < 100; i += THREADS) {
    sm[OFF_SB1 + i] = p.svb1[i];
    sm[OFF_SW3 + i] = p.svW3[i];
  }
  for (int i = tid; i < 112; i += THREADS) sm[OFF_SB2 + i] = (i < 100) ? p.svb2[i] : 0.f;
  for (int i = tid; i < 97; i += THREADS) sm[OFF_TG + i] = p.tg[i];
  if (tid < NSTK) sm[OFF_STR + tid] = p.strikes[tid];
  if (tid < 20) {
    sm[OFF_VDW1 + tid] = p.vdW1[tid];
    sm[OFF_VDB1 + tid] = p.vdb1[tid];
    sm[OFF_VDW2 + tid] = p.vdW2[tid];
    sm[OFF_VVW1 + tid] = p.vvW1[tid];
    sm[OFF_VVB1 + tid] = p.vvb1[tid];
    sm[OFF_VVW2 + tid] = p.vvW2[tid];
  }
  for (int i = tid; i < 160; i += THREADS) sm[OFF_VHW1 + i] = p.vhW1[i];
  for (int i = tid; i < 80; i += THREADS) sm[OFF_VHB1 + i] = p.vhb1[i];
  for (int i = tid; i < 1680; i += THREADS) sm[OFF_VHW2 + i] = p.vhW2[i];
  for (int i = tid; i < 84; i += THREADS) sm[OFF_VHB2 + i] = p.vhb2[i];
  __syncthreads();

  const float rate = p.rate[0];
  const float rho_t = tanhf(p.rho[0]);
  const float beta = sqrtf(fmaxf(1.f - rho_t * rho_t, 0.f));
  const float svb3 = p.svb3[0], vdb2 = p.vdb2[0], vvb2 = p.vvb2[0];
  int mats[4];
#pragma unroll
  for (int k = 0; k < 4; ++k) mats[k] = p.mats[k];

  const int path = tid >> 2; // 0..63
  const int kmat = tid & 3;  // maturity slice owned by this thread
  const int gpath = bid * PATHS_PER_BLK + path;

  float Slog = logf(p.S0[0]);
  float V = 0.5f / (1.f + expf(-p.v0[0]));
  float cv[NSTK];
#pragma unroll
  for (int s = 0; s < NSTK; ++s) cv[s] = 0.f;

  const int wave = tid >> 5, lane = tid & 31;
  const int row = lane & 15, hi = lane >> 4;

  for (int step = 0; step < NT; ++step) {
    const float t0 = sm[OFF_TG + step];
    const float t1 = sm[OFF_TG + step + 1];
    const float h = t1 - t0;
    const float sqh = sqrtf(h);
    const int day = __float2int_rn(t1 * 365.f);
    int idx = 0;
    bool im = false;
#pragma unroll
    for (int k = 3; k >= 0; --k) {
      if (day <= mats[k]) idx = k;
      im = im || (day == mats[k]);
    }

    // ---- Phase A: h1 = relu([t0,Slog,V] @ sv_W1 + b1) -> LDS (64 x 100, stride 104)
    {
      const int jbase = kmat * 25;
      float* h1row = &sm[OFF_H1 + path * 104];
#pragma unroll 5
      for (int jj = 0; jj < 25; ++jj) {
        const int j = jbase + jj;
        float a = fmaf(t0, sm[OFF_SW1 + j],
                  fmaf(Slog, sm[OFF_SW1 + 100 + j],
                  fmaf(V, sm[OFF_SW1 + 200 + j], sm[OFF_SB1 + j])));
        h1row[j] = fmaxf(a, 0.f);
      }
    }
    __syncthreads();

    // ---- Phase B: h2 = relu(h1 @ sv_W2 + b2) via V_WMMA_F32_16X16X4_F32
    // 64x112 output = 4 M-tiles x 7 N-tiles, K = 100 = 25 chunks of 4.
    for (int t = wave; t < 28; t += 8) {
      const int mt = t & 3, nt = t >> 2;
      v8f acc = {};
      const float* arow = &sm[OFF_H1 + (mt * 16 + row) * 104 + 2 * hi];
      const float* bcol = &sm[OFF_W2 + (2 * hi) * 112 + nt * 16 + row];
      for (int kc = 0; kc < 25; ++kc) {
        v2f a = *(const v2f*)(arow + 4 * kc); // (m, 4kc+2hi), (m, 4kc+2hi+1)
        v2f b;
        b.x = bcol[(4 * kc) * 112];     // (4kc+2hi,   n)
        b.y = bcol[(4 * kc + 1) * 112]; // (4kc+2hi+1, n)
        acc = __builtin_amdgcn_wmma_f32_16x16x4_f32(false, a, false, b,
                                                    (short)0, acc, false, false);
      }
      const int n = nt * 16 + row;
      const float bias = sm[OFF_SB2 + n];
#pragma unroll
      for (int r = 0; r < 8; ++r) {
        const int m = mt * 16 + hi * 8 + r;
        sm[OFF_H2 + m * 112 + n] = fmaxf(acc[r] + bias, 0.f);
      }
    }
    __syncthreads();

    // ---- Phase C: per-(path, maturity) scalar work ----
    const float* h2row = &sm[OFF_H2 + path * 112];
    float accp = svb3;
    for (int j = 0; j < 100; ++j) accp = fmaf(h2row[j], sm[OFF_SW3 + j], accp);
    const float pd = softplusf(accp);

    float vd = vdb2, vvs = vvb2;
#pragma unroll 4
    for (int o = 0; o < 20; ++o) {
      vd = fmaf(fmaxf(fmaf(V, sm[OFF_VDW1 + o], sm[OFF_VDB1 + o]), 0.f), sm[OFF_VDW2 + o], vd);
      vvs = fmaf(fmaxf(fmaf(V, sm[OFF_VVW1 + o], sm[OFF_VVB1 + o]), 0.f), sm[OFF_VVW2 + o], vvs);
    }
    const float vv = softplusf(vvs);

    const float dW = sqh * p.z[gpath * NT + step];
    const float dB = rho_t * dW + beta * sqh * p.zz[gpath * NT + step];
    const float Vnew = V + vd * h + vv * dB;

    const float drift = rate - 0.5f * pd * pd;
    const float diff = pd * dW;
    const float drift_c = 1.f + fabsf(drift) * sqh;
    const float diff_c = 1.f + fabsf(pd) * sqh;
    const float Slog_new = Slog + drift * h / drift_c + diff / diff_c;
    const float S_new = expf(Slog_new);
    const float S_old = expf(Slog);
    const float disc1 = expf(-rate * t1);
    const float disc0 = expf(-rate * t0);
    const float dS = disc1 * S_new - disc0 * S_old;

    // hedge net for this thread's maturity (input [t0, Slog_old])
    float hh[20];
#pragma unroll
    for (int o = 0; o < 20; ++o) {
      hh[o] = fmaxf(fmaf(t0, sm[OFF_VHW1 + (kmat * 2 + 0) * 20 + o],
                    fmaf(Slog, sm[OFF_VHW1 + (kmat * 2 + 1) * 20 + o],
                         sm[OFF_VHB1 + kmat * 20 + o])), 0.f);
    }
    const float mask = (kmat >= idx) ? 1.f : 0.f;
    float price[NSTK];
#pragma unroll
    for (int s = 0; s < NSTK; ++s) {
      float hv = sm[OFF_VHB2 + kmat * NSTK + s];
#pragma unroll
      for (int o = 0; o < 20; ++o)
        hv = fmaf(hh[o], sm[OFF_VHW2 + (kmat * 20 + o) * NSTK + s], hv);
      cv[s] = fmaf(mask * softplusf(hv), dS, cv[s]);
      price[s] = disc1 * fmaxf(S_new - sm[OFF_STR + s], 0.f) - cv[s];
    }

    // ---- maturity step: deterministic block reduction (reuse H1 region) ----
    if (im) {
      if (kmat == idx) {
        float* pr = &sm[OFF_H1 + path * NSTK];
#pragma unroll
        for (int s = 0; s < NSTK; ++s) pr[s] = price[s];
      }
      __syncthreads();
      if (tid < NSTK) {
        float ssum = 0.f, ssq = 0.f;
        for (int pp = 0; pp < PATHS_PER_BLK; ++pp) {
          const float v = sm[OFF_H1 + pp * NSTK + tid];
          ssum += v;
          ssq = fmaf(v, v, ssq);
        }
        p.psum[(idx * NBLK + bid) * NSTK + tid] = ssum;
        p.psq[(idx * NBLK + bid) * NSTK + tid] = ssq;
      }
      __syncthreads();
    }

    Slog = Slog_new;
    V = Vnew;
  }
}

__global__ void lsv_finalize(const float* __restrict__ psum,
                             const float* __restrict__ psq,
                             float* __restrict__ out) {
  const int tid = threadIdx.x;
  if (tid >= NMAT * NSTK) return;
  float sum = 0.f, sq = 0.f;
  for (int b = 0; b < NBLK; ++b) {
    sum += psum[(tid / NSTK * NBLK + b) * NSTK + (tid % NSTK)];
    sq += psq[(tid / NSTK * NBLK + b) * NSTK + (tid % NSTK)];
  }
  const float mean = sum / (float)MC;
  const float var = (sq - sum * sum / (float)MC) / (float)(MC - 1);
  out[tid] = mean;                // pv_h  (4 x 21)
  out[NMAT * NSTK + tid] = var;   // pv_var (4 x 21)
}

extern "C" void kernel_launch(void* const* d_in, const int* in_sizes, int n_in,
                              void* d_out, int out_size, void* d_ws, size_t ws_size,
                              hipStream_t stream) {
  (void)in_sizes; (void)n_in; (void)out_size; (void)ws_size;
  P p;
  p.S0 = (const float*)d_in[0];
  p.rate = (const float*)d_in[1];
  p.z = (const float*)d_in[2];
  p.zz = (const float*)d_in[3];
  p.tg = (const float*)d_in[4];
  p.strikes = (const float*)d_in[5];
  p.v0 = (const float*)d_in[6];
  p.rho = (const float*)d_in[7];
  p.svW1 = (const float*)d_in[8];
  p.svb1 = (const float*)d_in[9];
  p.svW2 = (const float*)d_in[10];
  p.svb2 = (const float*)d_in[11];
  p.svW3 = (const float*)d_in[12];
  p.svb3 = (const float*)d_in[13];
  p.vhW1 = (const float*)d_in[14];
  p.vhb1 = (const float*)d_in[15];
  p.vhW2 = (const float*)d_in[16];
  p.vhb2 = (const float*)d_in[17];
  p.vdW1 = (const float*)d_in[18];
  p.vdb1 = (const float*)d_in[19];
  p.vdW2 = (const float*)d_in[20];
  p.vdb2 = (const float*)d_in[21];
  p.vvW1 = (const float*)d_in[22];
  p.vvb1 = (const float*)d_in[23];
  p.vvW2 = (const float*)d_in[24];
  p.vvb2 = (const float*)d_in[25];
  p.mats = (const int*)d_in[26];
  float* psum = (float*)d_ws;                    // 4*256*21 floats
  float* psq = psum + NMAT * NBLK * NSTK;        // 4*256*21 floats
  p.psum = psum;
  p.psq = psq;

  lsv_sim<<<NBLK, THREADS, SMEM_FLOATS * sizeof(float), stream>>>(p);
  lsv_finalize<<<1, 128, 0, stream>>>(psum, psq, (float*)d_out);
}